// SpatialRelation_5454608466450
// MI455X (gfx1250) — compile-verified
//
#include <hip/hip_runtime.h>
#include <math.h>

#define B_ 32
#define N_ 512
#define E_ 2048
#define D_ 128

typedef __attribute__((ext_vector_type(2))) float v2f;
typedef __attribute__((ext_vector_type(4))) float v4f;
typedef __attribute__((ext_vector_type(8))) float v8f;

// D = A(16x4 f32) * B(4x16 f32) + C(16x16 f32), wave32 WMMA
__device__ __forceinline__ v8f wmma4(v2f a, v2f b, v8f c) {
  return __builtin_amdgcn_wmma_f32_16x16x4_f32(false, a, false, b, (short)0, c, false, false);
}

__device__ __forceinline__ float sigmoid_f(float x) { return 1.0f / (1.0f + __expf(-x)); }
__device__ __forceinline__ float tanh_f(float x) {
  float t = __expf(-2.0f * fabsf(x));
  float r = (1.0f - t) / (1.0f + t);
  return copysignf(r, x);
}

// --------------------------------------------------------------------------
// Async global->LDS (CDNA5 GLOBAL_LOAD_ASYNC_TO_LDS_B128, ASYNCcnt-tracked),
// with a safe fallback if the builtin is not declared by this toolchain.
// Probe result: param 0 is `__device__ int __vector_size__(16) *` (AS1 int4*).
// --------------------------------------------------------------------------
#if defined(__has_builtin)
#if __has_builtin(__builtin_amdgcn_global_load_async_to_lds_b128) && \
    __has_builtin(__builtin_amdgcn_s_wait_asynccnt)
#define USE_ASYNC_LDS 1
#endif
#endif

#ifdef USE_ASYNC_LDS
typedef int vi4_t __attribute__((vector_size(16)));
typedef __attribute__((address_space(1))) vi4_t* g_vi4_p;
typedef __attribute__((address_space(3))) vi4_t* l_vi4_p;
#endif

__device__ __forceinline__ void cp16_to_lds(const float* g, float* l) {
#ifdef USE_ASYNC_LDS
  // LDS generic pointer: low 32 bits are the LDS byte offset (ISA 10.2).
  __builtin_amdgcn_global_load_async_to_lds_b128(
      (g_vi4_p)(uintptr_t)g, (l_vi4_p)(unsigned)(uintptr_t)l, 0, 0);
#else
  *(v4f*)l = *(const v4f*)g;
#endif
}
__device__ __forceinline__ void stage_fence() {
#ifdef USE_ASYNC_LDS
  __builtin_amdgcn_s_wait_asynccnt(0);
#endif
  __syncthreads();
}

// ---------------------------------------------------------------------------
// Y[M x 128] = act(X[M x 128] @ W[128 x 128] + bias); 64 rows per block.
// W staged in LDS as k-pair-interleaved float2 so each B fragment is one
// aligned ds_load_b64. ACT: 0 = identity, 1 = tanh
// ---------------------------------------------------------------------------
template <int ACT>
__global__ __launch_bounds__(256) void k_gemm128(const float* __restrict__ X,
                                                 const float* __restrict__ W,
                                                 const float* __restrict__ bias,
                                                 float* __restrict__ Y) {
  constexpr int LDX = 68;     // [64 x 64] chunk of X, row stride (words)
  constexpr int PAIRW = 288;  // W pair-row stride (words); 288 % 64 == 32
  __shared__ float Xs[64 * LDX];
  __shared__ float Ws2[32 * PAIRW];  // 32 k-pairs x 128 float2 (padded)

  const int tid = threadIdx.x;
  const int wave = tid >> 5;
  const int lane = tid & 31;
  const int l16 = lane & 15;
  const int khalf = lane >> 4;

  const int rowBase = blockIdx.x * 64;
  const int mTile = wave >> 1;  // 0..3
  const int nHalf = wave & 1;   // 0..1 (64 cols each)

  v8f acc[4] = {};

  for (int kc = 0; kc < 128; kc += 64) {
#pragma unroll
    for (int i = 0; i < 4; ++i) {  // X chunk: 64x64 (async to LDS)
      int idx = tid + i * 256;
      int r = idx >> 4, q = idx & 15;
      cp16_to_lds(X + (size_t)(rowBase + r) * D_ + kc + q * 4,
                  &Xs[r * LDX + q * 4]);
    }
#pragma unroll
    for (int i = 0; i < 8; ++i) {  // W chunk: 64x128 -> pair layout
      int idx = tid + i * 256;
      int r = idx >> 5, q = idx & 31;
      v4f v = *(const v4f*)(W + (size_t)(kc + r) * D_ + q * 4);
      float* dst = &Ws2[(r >> 1) * PAIRW + (r & 1)];
      dst[(q * 4 + 0) * 2] = v.x;
      dst[(q * 4 + 1) * 2] = v.y;
      dst[(q * 4 + 2) * 2] = v.z;
      dst[(q * 4 + 3) * 2] = v.w;
    }
    stage_fence();
#pragma unroll
    for (int k = 0; k < 64; k += 4) {
      int ka = k + 2 * khalf;  // even
      v2f a = *(const v2f*)(&Xs[(mTile * 16 + l16) * LDX + ka]);
#pragma unroll
      for (int t = 0; t < 4; ++t) {
        int n = nHalf * 64 + t * 16 + l16;
        v2f b = *(const v2f*)(&Ws2[(ka >> 1) * PAIRW + 2 * n]);
        acc[t] = wmma4(a, b, acc[t]);
      }
    }
    __syncthreads();
  }
#pragma unroll
  for (int t = 0; t < 4; ++t) {
    int n = nHalf * 64 + t * 16 + l16;
    float bv = bias[n];
#pragma unroll
    for (int i = 0; i < 8; ++i) {
      int r = rowBase + mTile * 16 + i + 8 * khalf;
      float v = acc[t][i] + bv;
      if (ACT == 1) v = tanh_f(v);
      Y[(size_t)r * D_ + n] = v;
    }
  }
}

// ---------------------------------------------------------------------------
// a_cur[b] = A[b](512x2048) @ Eg[b](2048x128); 64 rows per block, K chunk 32.
// ---------------------------------------------------------------------------
__global__ __launch_bounds__(256) void k_einsum(const float* __restrict__ A,
                                                const float* __restrict__ Eg,
                                                float* __restrict__ Yc) {
  constexpr int LDA = 36;     // [64 x 32]
  constexpr int PAIRW = 288;  // Eg pair-row stride (words)
  __shared__ float As[64 * LDA];
  __shared__ float Bs2[16 * PAIRW];  // 16 k-pairs x 128 float2

  const int tid = threadIdx.x;
  const int wave = tid >> 5;
  const int lane = tid & 31;
  const int l16 = lane & 15;
  const int khalf = lane >> 4;

  const int b = blockIdx.y;
  const int rowBase = blockIdx.x * 64;

  const float* Ab = A + (size_t)b * N_ * E_;
  const float* Eb = Eg + (size_t)b * E_ * D_;

  const int mTile = wave >> 1;
  const int nHalf = wave & 1;

  v8f acc[4] = {};

  for (int kc = 0; kc < E_; kc += 32) {
#pragma unroll
    for (int i = 0; i < 2; ++i) {  // A chunk 64x32 (async to LDS)
      int idx = tid + i * 256;
      int r = idx >> 3, q = idx & 7;
      cp16_to_lds(Ab + (size_t)(rowBase + r) * E_ + kc + q * 4,
                  &As[r * LDA + q * 4]);
    }
#pragma unroll
    for (int i = 0; i < 4; ++i) {  // Eg chunk 32x128 -> pair layout
      int idx = tid + i * 256;
      int r = idx >> 5, q = idx & 31;
      v4f v = *(const v4f*)(Eb + (size_t)(kc + r) * D_ + q * 4);
      float* dst = &Bs2[(r >> 1) * PAIRW + (r & 1)];
      dst[(q * 4 + 0) * 2] = v.x;
      dst[(q * 4 + 1) * 2] = v.y;
      dst[(q * 4 + 2) * 2] = v.z;
      dst[(q * 4 + 3) * 2] = v.w;
    }
    if (kc + 32 < E_) {  // prefetch next chunk (global_prefetch_b8)
      __builtin_prefetch(Eb + (size_t)(kc + 32 + (tid & 31)) * D_ + (tid >> 5) * 32, 0, 1);
      __builtin_prefetch(Ab + (size_t)(rowBase + (tid >> 2)) * E_ + kc + 32 + (tid & 3) * 8, 0, 1);
    }
    stage_fence();
#pragma unroll
    for (int k = 0; k < 32; k += 4) {
      int ka = k + 2 * khalf;
      v2f a = *(const v2f*)(&As[(mTile * 16 + l16) * LDA + ka]);
#pragma unroll
      for (int t = 0; t < 4; ++t) {
        int n = nHalf * 64 + t * 16 + l16;
        v2f bb = *(const v2f*)(&Bs2[(ka >> 1) * PAIRW + 2 * n]);
        acc[t] = wmma4(a, bb, acc[t]);
      }
    }
    __syncthreads();
  }
#pragma unroll
  for (int t = 0; t < 4; ++t) {
    int n = nHalf * 64 + t * 16 + l16;
#pragma unroll
    for (int i = 0; i < 8; ++i) {
      int r = rowBase + mTile * 16 + i + 8 * khalf;
      Yc[((size_t)b * N_ + r) * D_ + n] = acc[t][i];
    }
  }
}

// ---------------------------------------------------------------------------
// GRU-style gated update; 32 rows per block; K = 256 (concat a_cur|prop).
// ---------------------------------------------------------------------------
__global__ __launch_bounds__(256) void k_gate(
    const float* __restrict__ Acur, const float* __restrict__ PropIn,
    const float* __restrict__ Wr, const float* __restrict__ br,
    const float* __restrict__ Wz, const float* __restrict__ bz,
    const float* __restrict__ Wh, const float* __restrict__ bh,
    float* __restrict__ PropOut) {
  constexpr int LDA = 264;  // [32 x 256]: cols 0-127 a_cur, 128-255 prop
  constexpr int LDR = 136;  // [32 x 128]: r * prop
  __shared__ float As[32 * LDA];
  __shared__ float RP[32 * LDR];

  const int tid = threadIdx.x;
  const int wave = tid >> 5;
  const int lane = tid & 31;
  const int l16 = lane & 15;
  const int khalf = lane >> 4;

  const size_t rowBase = (size_t)blockIdx.x * 32;
  const int mTile = wave >> 2;  // 0..1
  const int quad = wave & 3;    // cols quad*32 .. +31

#pragma unroll
  for (int i = 0; i < 4; ++i) {
    int idx = tid + i * 256;
    int r = idx >> 5, q = idx & 31;
    cp16_to_lds(Acur + (rowBase + r) * D_ + q * 4, &As[r * LDA + q * 4]);
    cp16_to_lds(PropIn + (rowBase + r) * D_ + q * 4, &As[r * LDA + 128 + q * 4]);
  }
  stage_fence();

  v8f racc[2] = {}, zacc[2] = {};
  for (int k = 0; k < 256; k += 4) {
    int ka = k + 2 * khalf;
    v2f a = *(const v2f*)(&As[(mTile * 16 + l16) * LDA + ka]);
#pragma unroll
    for (int t = 0; t < 2; ++t) {
      int n = quad * 32 + t * 16 + l16;
      v2f wr, wz;
      wr.x = Wr[(size_t)ka * D_ + n];
      wr.y = Wr[(size_t)(ka + 1) * D_ + n];
      racc[t] = wmma4(a, wr, racc[t]);
      wz.x = Wz[(size_t)ka * D_ + n];
      wz.y = Wz[(size_t)(ka + 1) * D_ + n];
      zacc[t] = wmma4(a, wz, zacc[t]);
    }
  }

  float zv[2][8];
#pragma unroll
  for (int t = 0; t < 2; ++t) {
    int n = quad * 32 + t * 16 + l16;
    float brv = br[n], bzv = bz[n];
#pragma unroll
    for (int i = 0; i < 8; ++i) {
      int r = mTile * 16 + i + 8 * khalf;
      float rv = sigmoid_f(racc[t][i] + brv);
      RP[r * LDR + n] = rv * As[r * LDA + 128 + n];
      zv[t][i] = sigmoid_f(zacc[t][i] + bzv);
    }
  }
  __syncthreads();

  v8f hacc[2] = {};
  for (int k = 0; k < 128; k += 4) {  // a_cur half
    int ka = k + 2 * khalf;
    v2f a = *(const v2f*)(&As[(mTile * 16 + l16) * LDA + ka]);
#pragma unroll
    for (int t = 0; t < 2; ++t) {
      int n = quad * 32 + t * 16 + l16;
      v2f wh;
      wh.x = Wh[(size_t)ka * D_ + n];
      wh.y = Wh[(size_t)(ka + 1) * D_ + n];
      hacc[t] = wmma4(a, wh, hacc[t]);
    }
  }
  for (int k = 128; k < 256; k += 4) {  // r*prop half
    int ka = k + 2 * khalf;
    v2f a = *(const v2f*)(&RP[(mTile * 16 + l16) * LDR + (ka - 128)]);
#pragma unroll
    for (int t = 0; t < 2; ++t) {
      int n = quad * 32 + t * 16 + l16;
      v2f wh;
      wh.x = Wh[(size_t)ka * D_ + n];
      wh.y = Wh[(size_t)(ka + 1) * D_ + n];
      hacc[t] = wmma4(a, wh, hacc[t]);
    }
  }

#pragma unroll
  for (int t = 0; t < 2; ++t) {
    int n = quad * 32 + t * 16 + l16;
    float bhv = bh[n];
#pragma unroll
    for (int i = 0; i < 8; ++i) {
      int r = mTile * 16 + i + 8 * khalf;
      float h = tanh_f(hacc[t][i] + bhv);
      float p = As[r * LDA + 128 + n];
      float z = zv[t][i];
      PropOut[(rowBase + r) * D_ + n] = (1.0f - z) * p + z * h;
    }
  }
}

// ---------------------------------------------------------------------------
// logits[row] = tanh(X[row]@W1 + b1) . W2 + b2 ; 64 rows per block
// ---------------------------------------------------------------------------
__global__ __launch_bounds__(256) void k_att(const float* __restrict__ X,
                                             const float* __restrict__ W1,
                                             const float* __restrict__ b1,
                                             const float* __restrict__ W2,
                                             const float* __restrict__ b2,
                                             float* __restrict__ logits) {
  constexpr int LDX = 68;
  constexpr int PAIRW = 288;
  __shared__ float Xs[64 * LDX];
  __shared__ float Ws2[32 * PAIRW];
  __shared__ float lsum[64];

  const int tid = threadIdx.x;
  const int wave = tid >> 5;
  const int lane = tid & 31;
  const int l16 = lane & 15;
  const int khalf = lane >> 4;

  const int rowBase = blockIdx.x * 64;
  const int mTile = wave >> 1;
  const int nHalf = wave & 1;

  if (tid < 64) lsum[tid] = 0.0f;

  v8f acc[4] = {};
  for (int kc = 0; kc < 128; kc += 64) {
#pragma unroll
    for (int i = 0; i < 4; ++i) {
      int idx = tid + i * 256;
      int r = idx >> 4, q = idx & 15;
      cp16_to_lds(X + (size_t)(rowBase + r) * D_ + kc + q * 4,
                  &Xs[r * LDX + q * 4]);
    }
#pragma unroll
    for (int i = 0; i < 8; ++i) {
      int idx = tid + i * 256;
      int r = idx >> 5, q = idx & 31;
      v4f v = *(const v4f*)(W1 + (size_t)(kc + r) * D_ + q * 4);
      float* dst = &Ws2[(r >> 1) * PAIRW + (r & 1)];
      dst[(q * 4 + 0) * 2] = v.x;
      dst[(q * 4 + 1) * 2] = v.y;
      dst[(q * 4 + 2) * 2] = v.z;
      dst[(q * 4 + 3) * 2] = v.w;
    }
    stage_fence();
#pragma unroll
    for (int k = 0; k < 64; k += 4) {
      int ka = k + 2 * khalf;
      v2f a = *(const v2f*)(&Xs[(mTile * 16 + l16) * LDX + ka]);
#pragma unroll
      for (int t = 0; t < 4; ++t) {
        int n = nHalf * 64 + t * 16 + l16;
        v2f b = *(const v2f*)(&Ws2[(ka >> 1) * PAIRW + 2 * n]);
        acc[t] = wmma4(a, b, acc[t]);
      }
    }
    __syncthreads();
  }
#pragma unroll
  for (int i = 0; i < 8; ++i) {
    float s = 0.0f;
#pragma unroll
    for (int t = 0; t < 4; ++t) {
      int n = nHalf * 64 + t * 16 + l16;
      s += tanh_f(acc[t][i] + b1[n]) * W2[n];
    }
    s += __shfl_xor(s, 1, 32);
    s += __shfl_xor(s, 2, 32);
    s += __shfl_xor(s, 4, 32);
    s += __shfl_xor(s, 8, 32);
    if (l16 == 0) {
      int r = mTile * 16 + i + 8 * khalf;
      atomicAdd(&lsum[r], s);  // ds_add_f32
    }
  }
  __syncthreads();
  if (tid < 64) logits[rowBase + tid] = lsum[tid] + b2[0];
}

// ---------------------------------------------------------------------------
// per batch: softmax over 512 logits; mul[b,:] = sum_n w_n * Out[b,n,:]
// ---------------------------------------------------------------------------
__global__ __launch_bounds__(256) void k_pool(const float* __restrict__ logits,
                                              const float* __restrict__ Out,
                                              float* __restrict__ mul) {
  const int b = blockIdx.x;
  const int tid = threadIdx.x;
  __shared__ float w[N_];
  __shared__ float red[256];

  float l0 = logits[b * N_ + tid];
  float l1 = logits[b * N_ + 256 + tid];
  red[tid] = fmaxf(l0, l1);
  __syncthreads();
  for (int s = 128; s > 0; s >>= 1) {
    if (tid < s) red[tid] = fmaxf(red[tid], red[tid + s]);
    __syncthreads();
  }
  float mx = red[0];
  __syncthreads();
  float e0 = __expf(l0 - mx), e1 = __expf(l1 - mx);
  red[tid] = e0 + e1;
  __syncthreads();
  for (int s = 128; s > 0; s >>= 1) {
    if (tid < s) red[tid] += red[tid + s];
    __syncthreads();
  }
  float inv = 1.0f / red[0];
  w[tid] = e0 * inv;
  w[256 + tid] = e1 * inv;
  __syncthreads();

  int d = tid & 127, h = tid >> 7;
  float s = 0.0f;
  for (int n = h * 256; n < h * 256 + 256; ++n)
    s += w[n] * Out[((size_t)b * N_ + n) * D_ + d];
  red[tid] = s;
  __syncthreads();
  if (tid < 128) mul[b * D_ + d] = red[tid] + red[tid + 128];
}

// ---------------------------------------------------------------------------
// residual mean over initial node+edge states; res = relu(tanh(mul)+residual)
// ---------------------------------------------------------------------------
__global__ void k_final(const float* __restrict__ prop0,
                        const float* __restrict__ edge0,
                        const float* __restrict__ mul,
                        float* __restrict__ res) {
  int b = blockIdx.x, d = threadIdx.x;  // 32 x 128
  const float* p = prop0 + (size_t)b * N_ * D_ + d;
  float s = 0.0f;
  for (int n = 0; n < N_; ++n) s += p[(size_t)n * D_];
  const float* e = edge0 + (size_t)b * E_ * D_ + d;
  for (int i = 0; i < E_; ++i) s += e[(size_t)i * D_];
  float residual = s * (1.0f / (float)(N_ + E_));
  float g = tanh_f(mul[b * D_ + d]);
  float v = g + residual;
  res[b * D_ + d] = v > 0.0f ? v : 0.0f;
}

extern "C" void kernel_launch(void* const* d_in, const int* in_sizes, int n_in,
                              void* d_out, int out_size, void* d_ws,
                              size_t ws_size, hipStream_t stream) {
  (void)in_sizes; (void)n_in; (void)out_size; (void)ws_size;
  const float* prop0 = (const float*)d_in[0];
  const float* edge0 = (const float*)d_in[1];
  const float* Amat = (const float*)d_in[2];
  // d_in[3] = tem_features (unused by reference)
  const float* link_W = (const float*)d_in[4];
  const float* link_b = (const float*)d_in[5];
  const float* reset_W = (const float*)d_in[6];
  const float* reset_b = (const float*)d_in[7];
  const float* update_W = (const float*)d_in[8];
  const float* update_b = (const float*)d_in[9];
  const float* trans_W = (const float*)d_in[10];
  const float* trans_b = (const float*)d_in[11];
  const float* att_W1 = (const float*)d_in[12];
  const float* att_b1 = (const float*)d_in[13];
  const float* att_W2 = (const float*)d_in[14];
  const float* att_b2 = (const float*)d_in[15];
  const float* out_W = (const float*)d_in[16];
  const float* out_b = (const float*)d_in[17];

  float* o_res = (float*)d_out;           // [B*D]
  float* o_mul = o_res + B_ * D_;         // [B*D]
  float* o_edge = o_mul + B_ * D_;        // [B*E*D] final edge states live here

  float* ws = (float*)d_ws;
  float* a_cur = ws;                              // B*N*D
  float* prop = a_cur + (size_t)B_ * N_ * D_;     // B*N*D
  float* outbuf = prop + (size_t)B_ * N_ * D_;    // B*N*D
  float* logits = outbuf + (size_t)B_ * N_ * D_;  // B*N

  for (int step = 0; step < 3; ++step) {
    const float* eIn = (step == 0) ? edge0 : o_edge;
    k_gemm128<0><<<dim3(B_ * E_ / 64), 256, 0, stream>>>(eIn, link_W, link_b, o_edge);
    k_einsum<<<dim3(N_ / 64, B_), 256, 0, stream>>>(Amat, o_edge, a_cur);
    const float* pIn = (step == 0) ? prop0 : prop;
    k_gate<<<dim3(B_ * N_ / 32), 256, 0, stream>>>(a_cur, pIn, reset_W, reset_b,
                                                   update_W, update_b, trans_W,
                                                   trans_b, prop);
  }
  k_att<<<dim3(B_ * N_ / 64), 256, 0, stream>>>(prop, att_W1, att_b1, att_W2,
                                                att_b2, logits);
  k_gemm128<1><<<dim3(B_ * N_ / 64), 256, 0, stream>>>(prop, out_W, out_b, outbuf);
  k_pool<<<dim3(B_), 256, 0, stream>>>(logits, outbuf, o_mul);
  k_final<<<dim3(B_), D_, 0, stream>>>(prop0, edge0, o_mul, o_res);
}